// Mol2MSLoss_38611755991898
// MI455X (gfx1250) — compile-verified
//
#include <hip/hip_runtime.h>
#include <hip/hip_bf16.h>
#include <math.h>

#define BATCH 32
#define SEQ   2048
#define NTILE (SEQ / 16)   // 128 column tiles per batch row

typedef float v2f __attribute__((ext_vector_type(2)));
typedef float v8f __attribute__((ext_vector_type(8)));

// Workspace layout (floats):
//   [0..31]   soft_intersect[b]
//   [32..63]  soft_union[b] (float-valued counts)
//   [64] sum (pred_mz-mz)^2        [65] sum (pred_int-int)^2
//   [66] max|mz|   (uint-bits via atomicMax; values >=0 so uint order == float order)
//   [67] max|int|  (uint-bits)
//   [68] sum |sign(pred_mz)-sign(mz)|   [69] sum |sign(pred_int)-sign(int)|

__device__ __forceinline__ float sgnf(float x) {
    return (x > 0.0f) ? 1.0f : ((x < 0.0f) ? -1.0f : 0.0f);
}

// Single-instruction running min (avoids llvm.minnum's sNaN canonicalize pair).
__device__ __forceinline__ void vmin(float& d, float s) {
    asm("v_min_num_f32 %0, %0, %1" : "+v"(d) : "v"(s));
}

__global__ void init_ws_kernel(float* ws) {
    int t = threadIdx.x;
    if (t < 128) ws[t] = 0.0f;
}

// ---- Elementwise reductions: MSE sums, max-abs, sign penalty, union counts ----
__global__ void __launch_bounds__(256)
reduce_kernel(const float* __restrict__ pred_mz, const float* __restrict__ mz,
              const float* __restrict__ intensity, const float* __restrict__ pred_intensity,
              float* __restrict__ ws) {
    const int gid  = blockIdx.x * blockDim.x + threadIdx.x;
    const int base = gid * 4;

    float s_mz = 0.f, s_in = 0.f, mx_mz = 0.f, mx_in = 0.f;
    float sg_mz = 0.f, sg_in = 0.f, uni = 0.f;
#pragma unroll
    for (int k = 0; k < 4; ++k) {
        const int e = base + k;
        const float pm  = pred_mz[e];
        const float m   = mz[e];
        const float it  = intensity[e];
        const float pit = pred_intensity[e];
        const float dm = pm - m;   s_mz += dm * dm;
        const float di = pit - it; s_in += di * di;
        mx_mz = fmaxf(mx_mz, fabsf(m));
        mx_in = fmaxf(mx_in, fabsf(it));
        sg_mz += fabsf(sgnf(pm)  - sgnf(m));
        sg_in += fabsf(sgnf(pit) - sgnf(it));
        uni += ((pm >= 0.0f) ? 1.0f : 0.0f) + ((m >= 0.0f) ? 1.0f : 0.0f);
    }

#pragma unroll
    for (int off = 16; off > 0; off >>= 1) {
        s_mz  += __shfl_xor(s_mz,  off, 32);
        s_in  += __shfl_xor(s_in,  off, 32);
        sg_mz += __shfl_xor(sg_mz, off, 32);
        sg_in += __shfl_xor(sg_in, off, 32);
        uni   += __shfl_xor(uni,   off, 32);
        mx_mz = fmaxf(mx_mz, __shfl_xor(mx_mz, off, 32));
        mx_in = fmaxf(mx_in, __shfl_xor(mx_in, off, 32));
    }

    if ((threadIdx.x & 31) == 0) {
        atomicAdd(&ws[64], s_mz);
        atomicAdd(&ws[65], s_in);
        atomicMax((unsigned int*)&ws[66], __float_as_uint(mx_mz));
        atomicMax((unsigned int*)&ws[67], __float_as_uint(mx_in));
        atomicAdd(&ws[68], sg_mz);
        atomicAdd(&ws[69], sg_in);
        const int b = base / SEQ;          // whole wave is inside one batch row
        atomicAdd(&ws[32 + b], uni);
    }
}

// ---- Soft-Jaccard pairwise kernel via V_WMMA_F32_16X16X4_F32 ----
// d2_ij = a_i^2 - 2 a_i b_j + b_j^2 as a K=4 bilinear form:
//   A row i = [a^2, -2a, 1, 0] ; B col j = [1, b, b^2, 0]
// Each wave owns TWO 16-row i-tiles (shared B per j-tile halves per-WMMA
// overhead) and loops over all 128 mz tiles staged in LDS.
// 256 blocks x 8 waves = 2048 waves = 32 batches * 64 tile-pairs: exact fit,
// no divergence before the WMMAs -> EXEC all-ones as the ISA requires.
__global__ void __launch_bounds__(256)
soft_jaccard_kernel(const float* __restrict__ pred_mz, const float* __restrict__ mz,
                    const float* __restrict__ pred_intensity, float* __restrict__ ws) {
    __shared__ float s_mz[SEQ];            // 8 KB: masked mz row for this batch

    const int lane  = threadIdx.x & 31;
    const int gwave = blockIdx.x * 8 + (threadIdx.x >> 5);
    const int b     = gwave >> 6;          // 64 tile-pairs per batch; 8 blocks/batch
    const int ti0   = (gwave & 63) * 2;    // first i-tile; second is ti0+1
    const int half  = lane >> 4;           // 0: lanes 0-15, 1: lanes 16-31
    const int l     = lane & 15;

    // Stage masked mz[b,:] into LDS (mask: mz>=0 ? mz : 0  ==  fmaxf(mz,0))
    const float* mrow = mz + b * SEQ;
    for (int i = threadIdx.x; i < SEQ; i += 256)
        s_mz[i] = fmaxf(mrow[i], 0.0f);
    __syncthreads();

    // A operands (32-bit A 16x4 layout): lanes 0-15 -> (K0,K1)=(a^2,-2a) for M=l,
    // lanes 16-31 -> (K2,K3)=(1,0)
    const float a0 = fmaxf(pred_mz[b * SEQ + ti0 * 16 + l], 0.0f);
    const float a1 = fmaxf(pred_mz[b * SEQ + (ti0 + 1) * 16 + l], 0.0f);
    v2f A0, A1;
    A0.x = half ? 1.0f : a0 * a0;
    A0.y = half ? 0.0f : -2.0f * a0;
    A1.x = half ? 1.0f : a1 * a1;
    A1.y = half ? 0.0f : -2.0f * a1;

    float dmin0[8], dmin1[8];
#pragma unroll
    for (int k = 0; k < 8; ++k) { dmin0[k] = 3.0e38f; dmin1[k] = 3.0e38f; }

    const v8f zeroC = {};
    for (int tj = 0; tj < NTILE; ++tj) {
        const float bv = s_mz[tj * 16 + l];
        // B operand (4x16): lanes 0-15 -> (K0,K1)=(1,b) for N=l, lanes 16-31 -> (b^2,0)
        v2f Bm;
        Bm.x = half ? bv * bv : 1.0f;
        Bm.y = half ? 0.0f    : bv;
        v8f D0 = __builtin_amdgcn_wmma_f32_16x16x4_f32(
            false, A0, false, Bm, (short)0, zeroC, false, false);
        v8f D1 = __builtin_amdgcn_wmma_f32_16x16x4_f32(
            false, A1, false, Bm, (short)0, zeroC, false, false);
#pragma unroll
        for (int k = 0; k < 8; ++k) { vmin(dmin0[k], D0[k]); vmin(dmin1[k], D1[k]); }
    }

    // Min across columns: C/D layout puts N=lane&15 per lane; reduce within each
    // 16-lane half (xor masks 1,2,4,8 never cross the half boundary).
#pragma unroll
    for (int off = 1; off < 16; off <<= 1) {
#pragma unroll
        for (int k = 0; k < 8; ++k) {
            vmin(dmin0[k], __shfl_xor(dmin0[k], off, 32));
            vmin(dmin1[k], __shfl_xor(dmin1[k], off, 32));
        }
    }

    // VGPR k holds row M=k (lanes 0-15) or M=k+8 (lanes 16-31).
    if (l == 0) {
        const float* pmrow = pred_mz        + b * SEQ + ti0 * 16;
        const float* pirow = pred_intensity + b * SEQ + ti0 * 16;
        float acc = 0.0f;
#pragma unroll
        for (int k = 0; k < 8; ++k) {
            const int m = k + half * 8;
            float d  = __fsqrt_rn(fmaxf(dmin0[k], 0.0f));
            float sm = __expf(-10.0f * d);               // exp(-min|a-b| / 0.1)
            float pi = (pmrow[m] >= 0.0f) ? pirow[m] : 0.0f;
            acc += sm * pi;
            d  = __fsqrt_rn(fmaxf(dmin1[k], 0.0f));
            sm = __expf(-10.0f * d);
            pi = (pmrow[16 + m] >= 0.0f) ? pirow[16 + m] : 0.0f;
            acc += sm * pi;
        }
        atomicAdd(&ws[b], acc);
    }
}

__global__ void finalize_kernel(const float* __restrict__ ws, float* __restrict__ out) {
    const int t = threadIdx.x;    // 32 threads
    float sj = 1.0f - (ws[t] + 1e-8f) / (ws[32 + t] + 1e-8f);
#pragma unroll
    for (int off = 16; off > 0; off >>= 1) sj += __shfl_xor(sj, off, 32);
    if (t == 0) {
        const float N = (float)(BATCH * SEQ);
        const float max_mz = __uint_as_float(((const unsigned int*)ws)[66]);
        const float max_in = __uint_as_float(((const unsigned int*)ws)[67]);
        out[0] = sj / (float)BATCH;
        out[1] = (ws[64] / N) / (max_mz * max_mz);
        out[2] = (ws[65] / N) / (max_in * max_in);
        out[3] = 0.5f * (ws[68] / N + ws[69] / N);
    }
}

extern "C" void kernel_launch(void* const* d_in, const int* in_sizes, int n_in,
                              void* d_out, int out_size, void* d_ws, size_t ws_size,
                              hipStream_t stream) {
    const float* pred_mz        = (const float*)d_in[0];
    const float* mz             = (const float*)d_in[1];
    const float* intensity      = (const float*)d_in[2];
    const float* pred_intensity = (const float*)d_in[3];
    float* out = (float*)d_out;
    float* ws  = (float*)d_ws;
    (void)in_sizes; (void)n_in; (void)out_size; (void)ws_size;

    hipLaunchKernelGGL(init_ws_kernel, dim3(1), dim3(128), 0, stream, ws);
    hipLaunchKernelGGL(reduce_kernel, dim3((BATCH * SEQ) / (256 * 4)), dim3(256), 0, stream,
                       pred_mz, mz, intensity, pred_intensity, ws);
    hipLaunchKernelGGL(soft_jaccard_kernel, dim3(BATCH * (NTILE / 2) / 8), dim3(256), 0, stream,
                       pred_mz, mz, pred_intensity, ws);
    hipLaunchKernelGGL(finalize_kernel, dim3(1), dim3(32), 0, stream, ws, out);
}